// WavLayer_54597624267184
// MI455X (gfx1250) — compile-verified
//
#include <hip/hip_runtime.h>
#include <hip/hip_bf16.h>

typedef __attribute__((ext_vector_type(2))) float v2f;
typedef __attribute__((ext_vector_type(8))) float v8f;

#define BATCH 16
#define S1 65536   // 256*256
#define S2 16384   // 128*128
#define HW 262144  // 512*512

// ---------------- DWT (db2, periodic, downsample at odd indices) ----------------
// out[m] = sum_k f[k] * x[(2m+1-k) mod N]; 2D separable: rows (axis -2) then cols.
__global__ void dwt2_kernel(const float* __restrict__ src,
                            float* __restrict__ oa,  float* __restrict__ oad,
                            float* __restrict__ oda, float* __restrict__ odd,
                            int N) {
    const int M = N >> 1;
    long long idx = (long long)blockIdx.x * blockDim.x + threadIdx.x;
    long long total = (long long)BATCH * M * M;
    if (idx >= total) return;
    int mc = (int)(idx % M);
    long long t = idx / M;
    int mr = (int)(t % M);
    int b  = (int)(t / M);

    const float lo[4] = {-0.12940952255092145f, 0.22414386804185735f,
                          0.836516303737469f,   0.48296291314469025f};
    const float hi[4] = {-0.48296291314469025f, 0.836516303737469f,
                         -0.22414386804185735f, -0.12940952255092145f};

    const float* xb = src + (size_t)b * N * N;
    const int rb = 2 * mr + 1, cb = 2 * mc + 1, msk = N - 1;

    float sa = 0.f, sad = 0.f, sda = 0.f, sdd = 0.f;
#pragma unroll
    for (int kr = 0; kr < 4; ++kr) {
        const int rr = (rb - kr) & msk;
        const float* row = xb + (size_t)rr * N;
        float rl = 0.f, rh = 0.f;
#pragma unroll
        for (int kc = 0; kc < 4; ++kc) {
            const float v = row[(cb - kc) & msk];
            rl += lo[kc] * v;
            rh += hi[kc] * v;
        }
        sa  += lo[kr] * rl;  sad += lo[kr] * rh;   // row-LP
        sda += hi[kr] * rl;  sdd += hi[kr] * rh;   // row-HP
    }
    size_t o = (size_t)b * M * M + (size_t)mr * M + mc;
    oa[o] = sa; oad[o] = sad; oda[o] = sda; odd[o] = sdd;
}

// ---------------- per-batch max(|x|) reductions ----------------
__global__ void zero_u32_kernel(unsigned* __restrict__ p, int n) {
    int i = blockIdx.x * blockDim.x + threadIdx.x;
    if (i < n) p[i] = 0u;
}

__global__ void maxabs_kernel(const float* __restrict__ src,
                              unsigned* __restrict__ dst, int perBatch) {
    const int b = blockIdx.y;
    const float* p = src + (size_t)b * perBatch;
    unsigned v = 0u;
    for (int i = blockIdx.x * blockDim.x + threadIdx.x; i < perBatch;
         i += gridDim.x * blockDim.x) {
        unsigned bits = __float_as_uint(fabsf(p[i]));  // non-negative: bit order == value order
        v = (bits > v) ? bits : v;
    }
    __shared__ unsigned red[256];
    red[threadIdx.x] = v;
    __syncthreads();
    for (int s = 128; s > 0; s >>= 1) {
        if (threadIdx.x < s) {
            unsigned o = red[threadIdx.x + s];
            if (o > red[threadIdx.x]) red[threadIdx.x] = o;
        }
        __syncthreads();
    }
    if (threadIdx.x == 0) atomicMax(&dst[b], red[0]);
}

// ---------------- assemble: mosaic*hfw (ch0) + bilinear low path (ch1) ----------------
__global__ void assemble_kernel(const float* __restrict__ a2,  const float* __restrict__ ad2,
                                const float* __restrict__ da2, const float* __restrict__ dd2,
                                const float* __restrict__ ad1, const float* __restrict__ da1,
                                const float* __restrict__ dd1, const unsigned* __restrict__ mx,
                                const float* __restrict__ hfw, const float* __restrict__ lfw,
                                float* __restrict__ inp) {
    long long idx = (long long)blockIdx.x * blockDim.x + threadIdx.x;
    if (idx >= (long long)BATCH * HW) return;
    const int w = (int)(idx & 511);
    const int h = (int)((idx >> 9) & 511);
    const int b = (int)(idx >> 18);

    // ----- channel 0: normalized wavelet mosaic * hfw -----
    float val;
    if (h < 256 && w < 256) {
        const size_t base2 = (size_t)b * S2;
        if (h < 128 && w < 128)
            val = a2 [base2 + (size_t)h * 128 + w]            / __uint_as_float(mx[0 * BATCH + b]);
        else if (h < 128)
            val = ad2[base2 + (size_t)h * 128 + (w - 128)]    / __uint_as_float(mx[1 * BATCH + b]);
        else if (w < 128)
            val = da2[base2 + (size_t)(h - 128) * 128 + w]    / __uint_as_float(mx[2 * BATCH + b]);
        else
            val = dd2[base2 + (size_t)(h - 128) * 128 + (w - 128)] / __uint_as_float(mx[3 * BATCH + b]);
    } else {
        const size_t base1 = (size_t)b * S1;
        if (h < 256)
            val = ad1[base1 + (size_t)h * 256 + (w - 256)]          / __uint_as_float(mx[4 * BATCH + b]);
        else if (w < 256)
            val = da1[base1 + (size_t)(h - 256) * 256 + w]          / __uint_as_float(mx[5 * BATCH + b]);
        else
            val = dd1[base1 + (size_t)(h - 256) * 256 + (w - 256)]  / __uint_as_float(mx[6 * BATCH + b]);
    }
    const float high = val * hfw[(size_t)h * 512 + w];

    // ----- channel 1: bilinear upsample of (a2/m0)*lfw, 128 -> 512 -----
    const float m0inv = 1.f / __uint_as_float(mx[0 * BATCH + b]);
    float sr = fminf(fmaxf(((float)h + 0.5f) * 0.25f - 0.5f, 0.f), 127.f);
    float sc = fminf(fmaxf(((float)w + 0.5f) * 0.25f - 0.5f, 0.f), 127.f);
    int r0 = (int)floorf(sr); int r1 = min(r0 + 1, 127); float wr = sr - (float)r0;
    int c0 = (int)floorf(sc); int c1 = min(c0 + 1, 127); float wc = sc - (float)c0;
    const float* ab = a2 + (size_t)b * S2;
    float v00 = ab[r0 * 128 + c0] * lfw[r0 * 128 + c0];
    float v01 = ab[r0 * 128 + c1] * lfw[r0 * 128 + c1];
    float v10 = ab[r1 * 128 + c0] * lfw[r1 * 128 + c0];
    float v11 = ab[r1 * 128 + c1] * lfw[r1 * 128 + c1];
    float low = ((v00 * (1.f - wc) + v01 * wc) * (1.f - wr) +
                 (v10 * (1.f - wc) + v11 * wc) * wr) * m0inv;

    float* ob = inp + (size_t)b * 2 * HW + (size_t)h * 512 + w;
    ob[0]  = high;
    ob[HW] = low;
}

// ---------------- 3x3 conv (2 in-ch, 16 out-ch) as f32 WMMA GEMM ----------------
// One wave computes a 16(out-ch) x 16(pixel) tile.
// GEMM: D[16x16] = W[16x18(pad 20)] x patch[18(pad 20)x16], 5 chained 16x16x4 f32 WMMAs.
__global__ void conv_wmma_kernel(const float* __restrict__ inp,
                                 const float* __restrict__ cw,
                                 const float* __restrict__ cb,
                                 float* __restrict__ out) {
    const int wave = blockIdx.x * (blockDim.x >> 5) + (threadIdx.x >> 5);
    const int lane = threadIdx.x & 31;
    const int half = lane >> 4;   // selects K pair within a 4-wide K chunk
    const int nq   = lane & 15;   // A: M (out channel); B/D: N (pixel)

    const int wt = wave & 31;           // 32 tiles of 16 pixels per row
    const int h  = (wave >> 5) & 511;
    const int b  = wave >> 14;          // 512*32 = 2^14 tiles per image
    const int w0 = wt << 4;

    v8f acc = {};
#pragma unroll
    for (int c = 0; c < 5; ++c) {
        const int k0 = c * 4 + half * 2;
        v2f A, Bm;
#pragma unroll
        for (int t = 0; t < 2; ++t) {
            const int k = k0 + t;       // 0..19; 18,19 are zero padding
            float wv = 0.f, pv = 0.f;
            if (k < 18) {
                wv = cw[nq * 18 + k];                       // conv_w[o=nq][k]
                const int ci = k / 9;
                const int rem = k - ci * 9;
                const int r = rem / 3, s = rem - r * 3;
                const int hh = h + r - 1;
                const int ww = w0 + nq + s - 1;
                if (hh >= 0 && hh < 512 && ww >= 0 && ww < 512)
                    pv = inp[((size_t)b * 2 + ci) * HW + (size_t)hh * 512 + ww];
            }
            A[t]  = wv;
            Bm[t] = pv;
        }
        // (neg_a, A, neg_b, B, c_mod, C, reuse_a, reuse_b)
        acc = __builtin_amdgcn_wmma_f32_16x16x4_f32(false, A, false, Bm,
                                                    (short)0, acc, false, false);
    }
    // D layout: VGPR r -> M = r + 8*half, N = nq
#pragma unroll
    for (int r = 0; r < 8; ++r) {
        const int o = r + half * 8;
        out[((size_t)b * 16 + o) * HW + (size_t)h * 512 + (w0 + nq)] = acc[r] + cb[o];
    }
}

extern "C" void kernel_launch(void* const* d_in, const int* in_sizes, int n_in,
                              void* d_out, int out_size, void* d_ws, size_t ws_size,
                              hipStream_t stream) {
    const float* x   = (const float*)d_in[0];  // (16,1,512,512)
    const float* hfw = (const float*)d_in[1];  // (512,512)
    const float* lfw = (const float*)d_in[2];  // (128,128)
    const float* cw  = (const float*)d_in[3];  // (16,2,3,3)
    const float* cb  = (const float*)d_in[4];  // (16,)
    float* out = (float*)d_out;                // (16,16,512,512)

    float* ws = (float*)d_ws;
    float* a1  = ws;                          // B*S1
    float* ad1 = a1  + (size_t)BATCH * S1;
    float* da1 = ad1 + (size_t)BATCH * S1;
    float* dd1 = da1 + (size_t)BATCH * S1;
    float* a2  = dd1 + (size_t)BATCH * S1;    // B*S2
    float* ad2 = a2  + (size_t)BATCH * S2;
    float* da2 = ad2 + (size_t)BATCH * S2;
    float* dd2 = da2 + (size_t)BATCH * S2;
    unsigned* mx = (unsigned*)(dd2 + (size_t)BATCH * S2);   // 7*B maxima
    float* inp = (float*)(mx + 128);          // (B,2,512,512)

    // level-1 DWT: 512 -> 256
    dwt2_kernel<<<(BATCH * S1 + 255) / 256, 256, 0, stream>>>(x, a1, ad1, da1, dd1, 512);
    // level-2 DWT: 256 -> 128
    dwt2_kernel<<<(BATCH * S2 + 255) / 256, 256, 0, stream>>>(a1, a2, ad2, da2, dd2, 256);

    // per-batch max(|.|) for the 7 coefficient blocks
    zero_u32_kernel<<<1, 128, 0, stream>>>(mx, 7 * BATCH);
    maxabs_kernel<<<dim3(8,  BATCH), 256, 0, stream>>>(a2,  mx + 0 * BATCH, S2);
    maxabs_kernel<<<dim3(8,  BATCH), 256, 0, stream>>>(ad2, mx + 1 * BATCH, S2);
    maxabs_kernel<<<dim3(8,  BATCH), 256, 0, stream>>>(da2, mx + 2 * BATCH, S2);
    maxabs_kernel<<<dim3(8,  BATCH), 256, 0, stream>>>(dd2, mx + 3 * BATCH, S2);
    maxabs_kernel<<<dim3(32, BATCH), 256, 0, stream>>>(ad1, mx + 4 * BATCH, S1);
    maxabs_kernel<<<dim3(32, BATCH), 256, 0, stream>>>(da1, mx + 5 * BATCH, S1);
    maxabs_kernel<<<dim3(32, BATCH), 256, 0, stream>>>(dd1, mx + 6 * BATCH, S1);

    // build 2-channel conv input
    assemble_kernel<<<(BATCH * HW + 255) / 256, 256, 0, stream>>>(
        a2, ad2, da2, dd2, ad1, da1, dd1, mx, hfw, lfw, inp);

    // WMMA conv: 16*512*32 = 262144 waves, 8 waves/block
    conv_wmma_kernel<<<(BATCH * 512 * 32) / 8, 256, 0, stream>>>(inp, cw, cb, out);
}